// DifferentISLoss_14714557956388
// MI455X (gfx1250) — compile-verified
//
#include <hip/hip_runtime.h>
#include <hip/hip_bf16.h>
#include <stdint.h>

typedef float v2f __attribute__((ext_vector_type(2)));
typedef float v8f __attribute__((ext_vector_type(8)));

#if defined(__gfx1250__) && __has_builtin(__builtin_amdgcn_wmma_f32_16x16x4_f32)
#define USE_WMMA 1
#else
#define USE_WMMA 0
#endif

#define WPB 8      // waves per block (wave32)

__device__ __forceinline__ void async_stage_row(const int* __restrict__ srcrow,
                                                unsigned ldsbase, int lane, int nbytes) {
  for (int off = lane * 16; off < nbytes; off += 32 * 16) {
    unsigned laddr = ldsbase + (unsigned)off;
    unsigned voff  = (unsigned)off;
    asm volatile("global_load_async_to_lds_b128 %0, %1, %2"
                 :: "v"(laddr), "v"(voff), "s"(srcrow) : "memory");
  }
#if __has_builtin(__builtin_amdgcn_s_wait_asynccnt)
  __builtin_amdgcn_s_wait_asynccnt(0);
#else
  asm volatile("s_wait_asynccnt 0" ::: "memory");
#endif
}

#if USE_WMMA
__device__ __forceinline__ v8f wmma_rowsum_acc(float x, float y, v8f c) {
  // D = A(16x4) x ones(4x16) + C : every C[m,n] accumulates the row-m sum of A.
  const v2f ones = {1.0f, 1.0f};
  v2f a = {x, y};
  return __builtin_amdgcn_wmma_f32_16x16x4_f32(
      false, a, false, ones, (short)0, c, false, false);
}
#endif

// Specialized: N (n_words) known at compile time, indices in [0, N), N power of two.
template <int N, int LOG2N>
__global__ __launch_bounds__(256) void k_sample_sums_fixed(
    const float* __restrict__ bigram,
    const float* __restrict__ startv,
    const float* __restrict__ endv,
    const float* __restrict__ bias,
    const int*  __restrict__ samples,
    float* __restrict__ outA,
    float* __restrict__ outB,
    int n_samples)
{
  __shared__ __align__(16) int smem[WPB * N];
  const int tid  = (int)threadIdx.x;
  const int wave = tid >> 5;
  const int lane = tid & 31;
  const int i = (int)blockIdx.x * WPB + wave;
  if (i >= n_samples) return;                 // wave-uniform exit

  // ---- async-stage this sample's index row (N int32) into LDS ----
  async_stage_row(samples + (size_t)i * (size_t)N,
                  (unsigned)(uintptr_t)(&smem[wave * N]), lane, N * 4);

  const int* srow = &smem[wave * N];
  const int npairs = N - 1;

#if USE_WMMA
  v8f accA = {0.f,0.f,0.f,0.f,0.f,0.f,0.f,0.f};
  v8f accH = {0.f,0.f,0.f,0.f,0.f,0.f,0.f,0.f};
#else
  float accAs = 0.f, accHs = 0.f;
#endif

  // ---- main loop: 128 pairs / iteration (4 consecutive pairs per lane) ----
  const int nfull = npairs >> 7;              // full 128-pair iterations
  for (int it = 0; it < nfull; ++it) {
    int j0 = (it << 7) + (lane << 2);
    int4 q = *(const int4*)(srow + j0);       // 16B-aligned ds_load_b128
    int e  = srow[j0 + 4];
    // 32-bit element offsets -> SADDR + 32-bit voffset addressing, shared by both tables
    unsigned i0 = ((unsigned)q.x << LOG2N) + (unsigned)q.y;
    unsigned i1 = ((unsigned)q.y << LOG2N) + (unsigned)q.z;
    unsigned i2 = ((unsigned)q.z << LOG2N) + (unsigned)q.w;
    unsigned i3 = ((unsigned)q.w << LOG2N) + (unsigned)e;
    float g0 = bigram[i0], g1 = bigram[i1], g2 = bigram[i2], g3 = bigram[i3];
    float h0 = bias[i0],   h1 = bias[i1],   h2 = bias[i2],   h3 = bias[i3];
#if USE_WMMA
    accA = wmma_rowsum_acc(g0, g1, accA);     // alternate chains to hide XDL latency
    accH = wmma_rowsum_acc(h0, h1, accH);
    accA = wmma_rowsum_acc(g2, g3, accA);
    accH = wmma_rowsum_acc(h2, h3, accH);
#else
    accAs += (g0 + g1) + (g2 + g3);
    accHs += (h0 + h1) + (h2 + h3);
#endif
  }

  // ---- masked tail (once per sample); EXEC stays all-ones ----
  {
    const int base = nfull << 7;
    int j0 = base + (lane << 2);
    float g[4], h[4];
#pragma unroll
    for (int k = 0; k < 4; ++k) {
      int jj = j0 + k;
      int cj = (jj < npairs) ? jj : (npairs - 1);
      unsigned idx = ((unsigned)srow[cj] << LOG2N) + (unsigned)srow[cj + 1];
      float gv = bigram[idx];
      float hv = bias[idx];
      bool ok = (jj < npairs);
      g[k] = ok ? gv : 0.f;
      h[k] = ok ? hv : 0.f;
    }
#if USE_WMMA
    accA = wmma_rowsum_acc(g[0], g[1], accA);
    accH = wmma_rowsum_acc(h[0], h[1], accH);
    accA = wmma_rowsum_acc(g[2], g[3], accA);
    accH = wmma_rowsum_acc(h[2], h[3], accH);
#else
    accAs += (g[0] + g[1]) + (g[2] + g[3]);
    accHs += (h[0] + h[1]) + (h[2] + h[3]);
#endif
  }

#if USE_WMMA
  // C layout: lane L (<16) holds C[0..7, L]; lane L+16 holds C[8..15, L].
  float tA = 0.f, tH = 0.f;
#pragma unroll
  for (int r = 0; r < 8; ++r) { tA += accA[r]; tH += accH[r]; }
  tA += __shfl_xor(tA, 16, 32);
  tH += __shfl_xor(tH, 16, 32);
#else
  float tA = accAs, tH = accHs;
#pragma unroll
  for (int m = 16; m > 0; m >>= 1) {
    tA += __shfl_xor(tA, m, 32);
    tH += __shfl_xor(tH, m, 32);
  }
#endif

  if (lane == 0) {
    float a = tA + startv[srow[0]] + endv[srow[N - 1]];
    outA[i] = a;        // unnormalized ratio A_i
    outB[i] = a + tH;   // B_i = A_i + sum(bias over pairs)
  }
}

// Generic fallback for arbitrary n (scalar accumulation, no LDS staging assumptions).
__global__ __launch_bounds__(256) void k_sample_sums_generic(
    const float* __restrict__ bigram,
    const float* __restrict__ startv,
    const float* __restrict__ endv,
    const float* __restrict__ bias,
    const int*  __restrict__ samples,
    float* __restrict__ outA,
    float* __restrict__ outB,
    int n, int n_samples)
{
  const int tid  = (int)threadIdx.x;
  const int wave = tid >> 5;
  const int lane = tid & 31;
  const int i = (int)blockIdx.x * WPB + wave;
  if (i >= n_samples) return;

  const int* srow = samples + (size_t)i * (size_t)n;
  const int npairs = n - 1;
  float accAs = 0.f, accHs = 0.f;
  for (int j = lane; j < npairs; j += 32) {
    unsigned idx = (unsigned)srow[j] * (unsigned)n + (unsigned)srow[j + 1];
    accAs += bigram[idx];
    accHs += bias[idx];
  }
  float tA = accAs, tH = accHs;
#pragma unroll
  for (int m = 16; m > 0; m >>= 1) {
    tA += __shfl_xor(tA, m, 32);
    tH += __shfl_xor(tH, m, 32);
  }
  if (lane == 0) {
    float a = tA + startv[srow[0]] + endv[srow[n - 1]];
    outA[i] = a;
    outB[i] = a + tH;
  }
}

__global__ __launch_bounds__(256) void k_finalize(
    const float* __restrict__ A, const float* __restrict__ B,
    const float* __restrict__ startv, const float* __restrict__ endv,
    const float* __restrict__ bigram, const float* __restrict__ bias,
    float* __restrict__ out, int n, int n_samples)
{
  const int tid = (int)threadIdx.x;
  float sA = 0.f, dot = 0.f, diag = 0.f;
  for (int idx = tid; idx < n_samples; idx += 256) {
    float a = A[idx];
    sA  += a;
    dot += a * B[idx];
  }
  for (int k = tid; k < n - 1; k += 256) {
    size_t o = (size_t)k * (size_t)n + (size_t)(k + 1);
    diag += bigram[o] + bias[o];
  }
  __shared__ float r0[256], r1[256], r2[256];
  r0[tid] = sA; r1[tid] = dot; r2[tid] = diag;
  __syncthreads();
  for (int off = 128; off > 0; off >>= 1) {
    if (tid < off) {
      r0[tid] += r0[tid + off];
      r1[tid] += r1[tid + off];
      r2[tid] += r2[tid + off];
    }
    __syncthreads();
  }
  if (tid == 0) {
    out[0] = -(startv[0] + endv[n - 1] + r2[0]) + r1[0] / r0[0];
  }
}

extern "C" void kernel_launch(void* const* d_in, const int* in_sizes, int n_in,
                              void* d_out, int out_size, void* d_ws, size_t ws_size,
                              hipStream_t stream) {
  const float* bigram  = (const float*)d_in[0];
  const float* startv  = (const float*)d_in[1];
  const float* endv    = (const float*)d_in[2];
  const float* bias    = (const float*)d_in[3];
  const int*   samples = (const int*)d_in[4];

  const int n         = in_sizes[1];            // n_words (2048)
  const int n_samples = in_sizes[4] / n;        // 8192

  float* dA = (float*)d_ws;
  float* dB = dA + n_samples;

  const int blocks = (n_samples + WPB - 1) / WPB;
  if (n == 2048) {
    k_sample_sums_fixed<2048, 11><<<blocks, 256, 0, stream>>>(
        bigram, startv, endv, bias, samples, dA, dB, n_samples);
  } else {
    k_sample_sums_generic<<<blocks, 256, 0, stream>>>(
        bigram, startv, endv, bias, samples, dA, dB, n, n_samples);
  }
  k_finalize<<<1, 256, 0, stream>>>(dA, dB, startv, endv, bigram, bias,
                                    (float*)d_out, n, n_samples);
}